// Encoder_Pos_52261162058072
// MI455X (gfx1250) — compile-verified
//
#include <hip/hip_runtime.h>

typedef __attribute__((ext_vector_type(2))) float v2f;
typedef __attribute__((ext_vector_type(8))) float v8f;

#define WMMA_F32X4(a, b, c) \
  __builtin_amdgcn_wmma_f32_16x16x4_f32(false, (a), false, (b), (short)0, (c), false, false)

#define NB   4      // batch
#define CH   256    // channels
#define HWD  32     // H == W
#define NP   1024   // H*W tokens
#define CQ   32     // C/8

// ---------------------------------------------------------------------------
// 1) Exact bilinear downsample by even factor s (mean of 2x2 block at
//    s*o + s/2 - 1), written PIXEL-MAJOR: dsT[b][p][ch] so conv K is contiguous.
//    ch is the fastest thread index -> coalesced writes.
// ---------------------------------------------------------------------------
__global__ void downsample_kernel(const float* __restrict__ in,
                                  float* __restrict__ dsT,
                                  int cshift, int s) {
  int idx = blockIdx.x * blockDim.x + threadIdx.x;   // [0, NB*NP*c)
  int c  = 1 << cshift;
  int ch = idx & (c - 1);
  int t2 = idx >> cshift;
  int p  = t2 & (NP - 1);
  int b  = t2 >> 10;
  int oy = p >> 5, ox = p & 31;
  int ih = s * oy + (s >> 1) - 1;
  int iw = s * ox + (s >> 1) - 1;
  int inHW = HWD * s;
  const float* base = in + (((long long)(b << cshift | ch)) * inHW + ih) * inHW + iw;
  dsT[idx] = 0.25f * (base[0] + base[1] + base[inHW] + base[inHW + 1]);
}

// ---------------------------------------------------------------------------
// 2) Pack conv weights [co][ci][kh][kw] -> wPack[tap][co][ci] (ci contiguous).
// ---------------------------------------------------------------------------
__global__ void pack_w_kernel(const float* __restrict__ w,
                              float* __restrict__ wp, int cin) {
  int idx = blockIdx.x * blockDim.x + threadIdx.x;   // [0, 9*CH*cin)
  int ci  = idx % cin;
  int t2  = idx / cin;
  int co  = t2 & (CH - 1);
  int tap = t2 >> 8;
  wp[idx] = w[((long long)(co * cin + ci)) * 9 + tap];
}

// ---------------------------------------------------------------------------
// 3) Transpose x [b][c][p] -> xT [b][p][c] (c fastest -> coalesced writes).
// ---------------------------------------------------------------------------
__global__ void transpose_x_kernel(const float* __restrict__ x,
                                   float* __restrict__ xT) {
  int idx = blockIdx.x * blockDim.x + threadIdx.x;   // [0, NB*NP*CH)
  int c = idx & (CH - 1);
  int p = (idx >> 8) & (NP - 1);
  int b = idx >> 18;
  xT[idx] = x[((long long)(b * CH + c)) * NP + p];
}

// ---------------------------------------------------------------------------
// 4) Conv3x3(SAME) + BN fold + ReLU accumulated into multiT[b][p][co].
//    Implicit GEMM, 32(M=co)x64(N=pix) register tile per wave, 8 accumulators.
//    All fragment loads are aligned float2 (K=ci contiguous): 6 b64 loads per
//    8 v_wmma, immediate-offset addressed.
// ---------------------------------------------------------------------------
__global__ void conv3x3_bn_relu_wmma(const float* __restrict__ dsT,   // [NB,NP,cin]
                                     const float* __restrict__ wPack, // [9,CH,cin]
                                     const float* __restrict__ sc,
                                     const float* __restrict__ bi,
                                     float* __restrict__ multiT,      // [NB,NP,CH]
                                     int cin, int first) {
  int wave = (blockIdx.x * blockDim.x + threadIdx.x) >> 5;  // [0, 512)
  int lane = threadIdx.x & 31;
  int half = lane >> 4, lm = lane & 15;

  int nt = wave & 15;          // 16 n-tiles of 64
  int mt = (wave >> 4) & 7;    // 8 m-tiles of 32
  int b  = wave >> 7;
  int m0 = mt * 32, n0 = nt * 64;

  int n[4], y[4], xcol[4];
#pragma unroll
  for (int j = 0; j < 4; ++j) {
    n[j] = n0 + 16 * j + lm;
    y[j] = n[j] >> 5;
    xcol[j] = n[j] & 31;
  }
  const float* dsb = dsT + (long long)b * NP * cin + 2 * half;     // + ka base
  const float* wr0 = wPack + (long long)(m0 + lm) * cin + 2 * half;
  const float* wr1 = wr0 + (long long)16 * cin;

  v8f acc[2][4] = {};
  for (int tap = 0; tap < 9; ++tap) {
    int kh = tap / 3, kw = tap % 3;
    float msk[4];
    const float* Bp[4];
#pragma unroll
    for (int j = 0; j < 4; ++j) {
      int iy = y[j] + kh - 1;
      int ix = xcol[j] + kw - 1;
      bool vok = (iy >= 0) & (iy < HWD) & (ix >= 0) & (ix < HWD);
      msk[j] = vok ? 1.0f : 0.0f;
      int off = vok ? (iy * HWD + ix) : 0;       // safe address when masked
      Bp[j] = dsb + (long long)off * cin;
    }
    const float* Ap0 = wr0 + (long long)tap * CH * cin;
    const float* Ap1 = wr1 + (long long)tap * CH * cin;
    for (int k0 = 0; k0 < cin; k0 += 4) {
      v2f a0 = *(const v2f*)(Ap0 + k0);
      v2f a1 = *(const v2f*)(Ap1 + k0);
      v2f bb[4];
#pragma unroll
      for (int j = 0; j < 4; ++j) {
        v2f t = *(const v2f*)(Bp[j] + k0);
        bb[j].x = t.x * msk[j];
        bb[j].y = t.y * msk[j];
      }
#pragma unroll
      for (int j = 0; j < 4; ++j) {
        acc[0][j] = WMMA_F32X4(a0, bb[j], acc[0][j]);
        acc[1][j] = WMMA_F32X4(a1, bb[j], acc[1][j]);
      }
    }
  }

  float* outp = multiT + (long long)b * NP * CH;
#pragma unroll
  for (int i = 0; i < 2; ++i) {
#pragma unroll
    for (int r = 0; r < 8; ++r) {
      int m = m0 + 16 * i + r + 8 * half;        // C-frag row mapping
      float s = sc[m], t = bi[m];
#pragma unroll
      for (int j = 0; j < 4; ++j) {
        float v = acc[i][j][r] * s + t;
        v = v > 0.0f ? v : 0.0f;
        long long idx = (long long)n[j] * CH + m;   // pixel-major
        if (first) outp[idx] = v; else outp[idx] += v;
      }
    }
  }
}

// ---------------------------------------------------------------------------
// 5) Generic batched GEMM, K ALWAYS CONTIGUOUS on both operands.
//    C[b, m*c_ms + n*c_ns] = alpha*sum_k A[b,m,k]*B[b,k,n] + bias[m] + res[...]
//    A element: A + b*a_bs + m*a_rs + k     (a_ks == 1)
//    B element: B + b*b_bs + n*b_ns + k     (b_ks == 1)
//    32x64 register tile per wave; 6 aligned b64 loads per 8 v_wmma.
// ---------------------------------------------------------------------------
__global__ void gemm_wmma(const float* __restrict__ A, long long a_bs, long long a_rs,
                          const float* __restrict__ Bm, long long b_bs, long long b_ns,
                          const float* __restrict__ bias,
                          const float* __restrict__ alpha,
                          const float* __restrict__ res,
                          float* __restrict__ Cm, long long c_bs, long long c_ms, long long c_ns,
                          int M, int Nn, int K) {
  int wave = (blockIdx.x * blockDim.x + threadIdx.x) >> 5;
  int lane = threadIdx.x & 31;
  int half = lane >> 4, lm = lane & 15;

  int ntiles = Nn >> 6, mtiles = M >> 5;
  int nt = wave % ntiles;
  int t2 = wave / ntiles;
  int mt = t2 % mtiles;
  int b  = t2 / mtiles;
  int m0 = mt * 32, n0 = nt * 64;

  const float* Ap0 = A + (long long)b * a_bs + (long long)(m0 + lm) * a_rs + 2 * half;
  const float* Ap1 = Ap0 + (long long)16 * a_rs;
  const float* Bp[4];
#pragma unroll
  for (int j = 0; j < 4; ++j)
    Bp[j] = Bm + (long long)b * b_bs + (long long)(n0 + 16 * j + lm) * b_ns + 2 * half;

  v8f acc[2][4] = {};
  for (int k0 = 0; k0 < K; k0 += 4) {
    v2f a0 = *(const v2f*)(Ap0 + k0);
    v2f a1 = *(const v2f*)(Ap1 + k0);
    v2f bb[4];
#pragma unroll
    for (int j = 0; j < 4; ++j) bb[j] = *(const v2f*)(Bp[j] + k0);
#pragma unroll
    for (int j = 0; j < 4; ++j) {
      acc[0][j] = WMMA_F32X4(a0, bb[j], acc[0][j]);
      acc[1][j] = WMMA_F32X4(a1, bb[j], acc[1][j]);
    }
  }

  float al = alpha ? alpha[0] : 1.0f;
#pragma unroll
  for (int i = 0; i < 2; ++i) {
#pragma unroll
    for (int r = 0; r < 8; ++r) {
      int m = m0 + 16 * i + r + 8 * half;
      float bs = bias ? bias[m] : 0.0f;
#pragma unroll
      for (int j = 0; j < 4; ++j) {
        int n = n0 + 16 * j + lm;
        long long idx = (long long)b * c_bs + (long long)m * c_ms + (long long)n * c_ns;
        float v = acc[i][j][r] * al + bs;
        if (res) v += res[idx];
        Cm[idx] = v;
      }
    }
  }
}

// ---------------------------------------------------------------------------
// 6) Fold position bias into kT[b][p][c]:  += rel_h[c,y] + rel_w[c,x]
//    (energy = q^T k + q^T pos == q^T (k + pos))
// ---------------------------------------------------------------------------
__global__ void add_pos_kernel(float* __restrict__ kT,
                               const float* __restrict__ rh,
                               const float* __restrict__ rw) {
  int idx = blockIdx.x * blockDim.x + threadIdx.x;   // NB*NP*CQ, c fastest
  int c = idx & (CQ - 1);
  int p = (idx >> 5) & (NP - 1);
  kT[idx] += rh[c * HWD + (p >> 5)] + rw[c * HWD + (p & 31)];
}

// ---------------------------------------------------------------------------
// 7) Row softmax: one 1024-wide row per wave32, shuffle reductions.
// ---------------------------------------------------------------------------
__global__ void softmax_kernel(float* __restrict__ attn) {
  int row  = (blockIdx.x * blockDim.x + threadIdx.x) >> 5;   // [0, NB*NP)
  int lane = threadIdx.x & 31;
  float* rp = attn + (long long)row * NP;

  float vals[32];
  float mx = -3.402823466e38f;
#pragma unroll
  for (int i = 0; i < 32; ++i) {
    vals[i] = rp[lane + i * 32];
    mx = fmaxf(mx, vals[i]);
  }
  for (int o = 16; o > 0; o >>= 1) mx = fmaxf(mx, __shfl_xor(mx, o, 32));

  float sum = 0.0f;
#pragma unroll
  for (int i = 0; i < 32; ++i) {
    vals[i] = __expf(vals[i] - mx);
    sum += vals[i];
  }
  for (int o = 16; o > 0; o >>= 1) sum += __shfl_xor(sum, o, 32);

  float inv = 1.0f / sum;
#pragma unroll
  for (int i = 0; i < 32; ++i) rp[lane + i * 32] = vals[i] * inv;
}

// ---------------------------------------------------------------------------
extern "C" void kernel_launch(void* const* d_in, const int* in_sizes, int n_in,
                              void* d_out, int out_size, void* d_ws, size_t ws_size,
                              hipStream_t stream) {
  const float* x  = (const float*)d_in[0];
  const float* xs[4] = {(const float*)d_in[1], (const float*)d_in[2],
                        (const float*)d_in[3], (const float*)d_in[4]};
  const float *w[4], *sc[4], *bi[4];
  for (int i = 0; i < 4; ++i) {
    w[i]  = (const float*)d_in[5 + 3 * i];
    sc[i] = (const float*)d_in[6 + 3 * i];
    bi[i] = (const float*)d_in[7 + 3 * i];
  }
  const float* qw = (const float*)d_in[17];
  const float* qb = (const float*)d_in[18];
  const float* kw = (const float*)d_in[19];
  const float* kb = (const float*)d_in[20];
  const float* vw = (const float*)d_in[21];
  const float* vb = (const float*)d_in[22];
  const float* rh = (const float*)d_in[23];
  const float* rw = (const float*)d_in[24];
  const float* gm = (const float*)d_in[25];

  // workspace layout (floats), ~20 MB total
  float* ws_f   = (float*)d_ws;
  float* ds_buf = ws_f;               // <= 4*1024*256   (dsT, pixel-major)
  float* multiT = ws_f + 1048576;     // 4*1024*256
  float* qT     = ws_f + 2097152;     // 4*1024*32
  float* kT     = ws_f + 2228224;     // 4*1024*32
  float* v_ws   = ws_f + 2359296;     // 4*256*1024 (row-major [c][p])
  float* xT     = ws_f + 3407872;     // 4*1024*256
  float* wPack  = ws_f + 4456448;     // <= 9*256*256

  float* out_f = (float*)d_out;       // [4,256,1024]
  float* attn  = out_f + 1048576;     // [4,1024,1024]

  const int cins[4]   = {32, 64, 128, 256};
  const int cshift[4] = {5, 6, 7, 8};
  const int scales[4] = {16, 8, 4, 2};

  // SPP branches: downsample(T) -> pack weights -> implicit-GEMM conv, summed.
  for (int i = 0; i < 4; ++i) {
    int total = NB * NP * cins[i];
    downsample_kernel<<<total / 256, 256, 0, stream>>>(xs[i], ds_buf, cshift[i], scales[i]);
    pack_w_kernel<<<(9 * CH * cins[i]) / 256, 256, 0, stream>>>(w[i], wPack, cins[i]);
    // waves = 4*8*16 = 512 -> 64 blocks of 8 waves
    conv3x3_bn_relu_wmma<<<64, 256, 0, stream>>>(ds_buf, wPack, sc[i], bi[i],
                                                 multiT, cins[i], i == 0);
  }

  // xT = transpose(x) once for the k/v projections.
  transpose_x_kernel<<<(NB * NP * CH) / 256, 256, 0, stream>>>(x, xT);

  // qT = (qw @ multi)^T + qb  (M=32,N=1024,K=256; 64 waves -> 8 blocks)
  gemm_wmma<<<8, 256, 0, stream>>>(qw, 0, CH,
                                   multiT, (long long)NP * CH, CH,
                                   qb, nullptr, nullptr,
                                   qT, (long long)NP * CQ, 1, CQ,   // transposed C
                                   CQ, NP, CH);
  // kT = (kw @ x)^T + kb
  gemm_wmma<<<8, 256, 0, stream>>>(kw, 0, CH,
                                   xT, (long long)NP * CH, CH,
                                   kb, nullptr, nullptr,
                                   kT, (long long)NP * CQ, 1, CQ,   // transposed C
                                   CQ, NP, CH);
  // kT += pos (fold rel_h + rel_w)
  add_pos_kernel<<<(NB * NP * CQ) / 256, 256, 0, stream>>>(kT, rh, rw);
  // v = vw @ x + vb, row-major [c][p]  (512 waves -> 64 blocks)
  gemm_wmma<<<64, 256, 0, stream>>>(vw, 0, CH,
                                    xT, (long long)NP * CH, CH,
                                    vb, nullptr, nullptr,
                                    v_ws, (long long)CH * NP, NP, 1,
                                    CH, NP, CH);
  // energy = q^T (k+pos): M=N=1024, K=32, both operands K-contiguous.
  // waves = 4*32*16 = 2048 -> 256 blocks.  Written straight into attn slice.
  gemm_wmma<<<256, 256, 0, stream>>>(qT, (long long)NP * CQ, CQ,
                                     kT, (long long)NP * CQ, CQ,
                                     nullptr, nullptr, nullptr,
                                     attn, (long long)NP * NP, NP, 1,
                                     NP, NP, CQ);
  // softmax rows in place (4096 rows -> 4096 waves -> 512 blocks)
  softmax_kernel<<<512, 256, 0, stream>>>(attn);
  // out = gamma*(v @ attn^T) + x : M=256,N=1024,K=1024 (both K-contiguous)
  gemm_wmma<<<64, 256, 0, stream>>>(v_ws, (long long)CH * NP, NP,
                                    attn, (long long)NP * NP, NP,
                                    nullptr, gm, x,
                                    out_f, (long long)CH * NP, NP, 1,
                                    CH, NP, NP);
}